// GCN2_68358699483283
// MI455X (gfx1250) — compile-verified
//
#include <hip/hip_runtime.h>
#include <math.h>

typedef float v2f __attribute__((ext_vector_type(2)));
typedef float v8f __attribute__((ext_vector_type(8)));

#define GCN_G   64
#define GCN_OUT 64
#define LDS_STRIDE 132   // 16 rows x 132 floats: conflict-free b64 reads across wave32

// ---------------------------------------------------------------- degree pass
__global__ void k_deg_init(float* __restrict__ deg, int n) {
    int i = blockIdx.x * blockDim.x + threadIdx.x;
    if (i < n) deg[i] = 1.0f;                      // self-loop contributes 1
}

__global__ void k_deg_count(const int* __restrict__ dst, float* __restrict__ deg, int e) {
    int i = blockIdx.x * blockDim.x + threadIdx.x;
    if (i < e) atomicAdd(&deg[dst[i]], 1.0f);
}

__global__ void k_deg_inv(float* __restrict__ deg, int n) {
    int i = blockIdx.x * blockDim.x + threadIdx.x;
    if (i < n) {
        float d = deg[i];
        deg[i] = (d > 0.0f) ? (1.0f / sqrtf(d)) : 0.0f;   // dinv
    }
}

// ------------------------------------------------------- WMMA fp32 GEMM H=X*W
// X: [nrows x K] row-major, W: [K x F] row-major, H: [nrows x F] row-major.
// Block = 128 threads = 4 waves. The block stages W[:, col0:col0+16] in LDS
// (transposed, padded stride) once; each wave computes one 16x16 output tile
// with full-precision V_WMMA_F32_16X16X4_F32, B-frags via single ds_load_b64.
__global__ void __launch_bounds__(128)
k_gemm_wmma_f32(const float* __restrict__ X, const float* __restrict__ W,
                float* __restrict__ H, int K, int F, int mtiles) {
    __shared__ float Wlds[16 * LDS_STRIDE];

    const int t     = threadIdx.x;
    const int wave  = t >> 5;
    const int lane  = t & 31;
    const int half  = lane >> 4;         // 0: lanes 0-15, 1: lanes 16-31
    const int l16   = lane & 15;
    const int col0  = blockIdx.y << 4;   // N tile (shared by all 4 waves)
    const int mtile = (blockIdx.x << 2) + wave;

    // Cooperative load: W[k][col0+n] -> Wlds[n*STRIDE + k]  (transposed)
    // K*16 elements, 128 threads -> K/8 elements each; coalesced over n.
    const int reps = (K << 4) >> 7;      // K*16/128
    for (int rep = 0; rep < reps; ++rep) {
        int idx = (rep << 7) + t;
        int k = idx >> 4;
        int n = idx & 15;
        Wlds[n * LDS_STRIDE + k] = W[(size_t)k * F + (col0 + n)];
    }
    __syncthreads();

    if (mtile < mtiles) {                // wave-uniform guard (EXEC stays all-1s)
        const int row0 = mtile << 4;
        // A-matrix 16x4 layout: lane -> M, (vgpr, lane-half) -> K pairs 0/1 vs 2/3
        const float* __restrict__ Xrow = X + (size_t)(row0 + l16) * K + (half << 1);
        // B-matrix 4x16 layout: lane -> N, (vgpr, lane-half) -> K (symmetric to A)
        const float* __restrict__ Wl = &Wlds[l16 * LDS_STRIDE + (half << 1)];

        v8f c = {};
#pragma unroll 8
        for (int k0 = 0; k0 < K; k0 += 4) {
            v2f a = *(const v2f*)(Xrow + k0);   // global_load_b64
            v2f b = *(const v2f*)(Wl + k0);     // ds_load_b64, bank-conflict-free
            // 8 args: (neg_a, A, neg_b, B, c_mod, C, reuse_a, reuse_b)
            c = __builtin_amdgcn_wmma_f32_16x16x4_f32(false, a, false, b,
                                                      (short)0, c, false, false);
        }

        // C/D layout: VGPR v holds M = v + 8*half, N = col0 + l16
        float* __restrict__ Ho = H + (size_t)(row0 + (half << 3)) * F + (col0 + l16);
#pragma unroll
        for (int v = 0; v < 8; ++v)
            Ho[(size_t)v * F] = c[v];
    }
}

// --------------------------------------------- aggregation: self-loop + bias
// out[i,f] = dinv[i]^2 * H[i,f] + bias[f]
__global__ void k_agg_init(const float* __restrict__ H, const float* __restrict__ dinv,
                           const float* __restrict__ bias, float* __restrict__ out,
                           int n, int fshift) {
    int idx = blockIdx.x * blockDim.x + threadIdx.x;
    int total = n << fshift;
    if (idx >= total) return;
    int i = idx >> fshift;
    int f = idx & ((1 << fshift) - 1);
    float di = dinv[i];
    out[idx] = di * di * H[idx] + bias[f];
}

// ------------------------------------------ aggregation: edge atomic scatter
// one thread per (edge, float4 chunk): out[dst] += dinv[s]*dinv[d] * H[src]
__global__ void k_edge_scatter(const int* __restrict__ src, const int* __restrict__ dst,
                               const float* __restrict__ dinv, const float* __restrict__ H,
                               float* __restrict__ out, int e, int fshift) {
    const int cshift = fshift - 2;                 // chunks per edge = F/4
    long long t = (long long)blockIdx.x * blockDim.x + threadIdx.x;
    long long total = (long long)e << cshift;
    if (t >= total) return;
    int ei = (int)(t >> cshift);
    int c  = ((int)t & ((1 << cshift) - 1)) << 2;  // starting feature of chunk
    int s = src[ei];
    int d = dst[ei];
    float w = dinv[s] * dinv[d];
    int F = 1 << fshift;
    const float4 h4 = *(const float4*)(H + (size_t)s * F + c);
    float* o = out + (size_t)d * F + c;
    atomicAdd(o + 0, w * h4.x);
    atomicAdd(o + 1, w * h4.y);
    atomicAdd(o + 2, w * h4.z);
    atomicAdd(o + 3, w * h4.w);
}

__global__ void k_relu(float* __restrict__ x, int total) {
    int i = blockIdx.x * blockDim.x + threadIdx.x;
    if (i < total) x[i] = fmaxf(x[i], 0.0f);
}

// ------------------------------------------------------------------- readout
__global__ void k_readout_zero(float* __restrict__ gsum, float* __restrict__ cnt) {
    int i = blockIdx.x * blockDim.x + threadIdx.x;
    if (i < GCN_G * GCN_OUT) gsum[i] = 0.0f;
    if (i < GCN_G) cnt[i] = 0.0f;
}

__global__ void k_readout_scatter(const float* __restrict__ h, const int* __restrict__ batch,
                                  float* __restrict__ gsum, float* __restrict__ cnt, int n) {
    int idx = blockIdx.x * blockDim.x + threadIdx.x;
    if (idx >= (n << 6)) return;                   // OUT = 64
    int i = idx >> 6;
    int f = idx & 63;
    int g = batch[i];
    atomicAdd(&gsum[(g << 6) + f], h[idx]);
    if (f == 0) atomicAdd(&cnt[g], 1.0f);
}

__global__ void k_readout_final(const float* __restrict__ gsum, const float* __restrict__ cnt,
                                float* __restrict__ outp) {
    int idx = blockIdx.x * blockDim.x + threadIdx.x;
    if (idx >= GCN_G * GCN_OUT) return;
    int g = idx >> 6;
    outp[idx] = gsum[idx] / fmaxf(cnt[g], 1.0f);
}

// ---------------------------------------------------------------------- host
extern "C" void kernel_launch(void* const* d_in, const int* in_sizes, int n_in,
                              void* d_out, int out_size, void* d_ws, size_t ws_size,
                              hipStream_t stream) {
    const float* x     = (const float*)d_in[0];    // [N,128]
    const int*   edge  = (const int*)d_in[1];      // [2,E] flat
    const int*   batch = (const int*)d_in[2];      // [N]
    const float* W1    = (const float*)d_in[3];
    const float* b1    = (const float*)d_in[4];
    const float* W2    = (const float*)d_in[5];
    const float* b2    = (const float*)d_in[6];
    const float* W3    = (const float*)d_in[7];
    const float* b3    = (const float*)d_in[8];

    const int IN = 128, HID = 128, OUTF = 64;
    const int N = in_sizes[0] / IN;                // 100000
    const int E = in_sizes[1] / 2;                 // 1600000
    const int* src = edge;
    const int* dst = edge + E;

    // workspace layout
    char* ws = (char*)d_ws;
    float* dinv = (float*)ws;                                  // N floats
    size_t off = ((size_t)N * sizeof(float) + 511) & ~(size_t)511;
    float* bufA = (float*)(ws + off);                          // N*128 (GEMM out)
    float* bufB = bufA + (size_t)N * HID;                      // N*128 (aggregated)
    float* gsum = bufB + (size_t)N * HID;                      // G*OUT
    float* cnt  = gsum + GCN_G * GCN_OUT;                      // G

    const int TB = 256;
    dim3 blk(TB);

    // ---- degree / normalization
    k_deg_init <<<dim3((N + TB - 1) / TB), blk, 0, stream>>>(dinv, N);
    k_deg_count<<<dim3((E + TB - 1) / TB), blk, 0, stream>>>(dst, dinv, E);
    k_deg_inv  <<<dim3((N + TB - 1) / TB), blk, 0, stream>>>(dinv, N);

    const int mt = N / 16;                         // 6250 M-tiles (N % 16 == 0)
    const int gx = (mt + 3) / 4;                   // 4 M-tiles per block (4 waves)
    long long tE128 = (long long)E * (HID / 4);
    long long tE64  = (long long)E * (OUTF / 4);
    int nblkE128 = (int)((tE128 + TB - 1) / TB);
    int nblkE64  = (int)((tE64 + TB - 1) / TB);

    // ---- layer 1: h1 = relu(agg(x @ W1) + b1)
    k_gemm_wmma_f32<<<dim3(gx, HID / 16), dim3(128), 0, stream>>>(x, W1, bufA, IN, HID, mt);
    k_agg_init     <<<dim3((N * HID + TB - 1) / TB), blk, 0, stream>>>(bufA, dinv, b1, bufB, N, 7);
    k_edge_scatter <<<dim3(nblkE128), blk, 0, stream>>>(src, dst, dinv, bufA, bufB, E, 7);
    k_relu         <<<dim3((N * HID + TB - 1) / TB), blk, 0, stream>>>(bufB, N * HID);

    // ---- layer 2: h2 = relu(agg(h1 @ W2) + b2)
    k_gemm_wmma_f32<<<dim3(gx, HID / 16), dim3(128), 0, stream>>>(bufB, W2, bufA, HID, HID, mt);
    k_agg_init     <<<dim3((N * HID + TB - 1) / TB), blk, 0, stream>>>(bufA, dinv, b2, bufB, N, 7);
    k_edge_scatter <<<dim3(nblkE128), blk, 0, stream>>>(src, dst, dinv, bufA, bufB, E, 7);
    k_relu         <<<dim3((N * HID + TB - 1) / TB), blk, 0, stream>>>(bufB, N * HID);

    // ---- layer 3: h3 = agg(h2 @ W3) + b3   (no relu)
    k_gemm_wmma_f32<<<dim3(gx, OUTF / 16), dim3(128), 0, stream>>>(bufB, W3, bufA, HID, OUTF, mt);
    k_agg_init     <<<dim3((N * OUTF + TB - 1) / TB), blk, 0, stream>>>(bufA, dinv, b3, bufB, N, 6);
    k_edge_scatter <<<dim3(nblkE64), blk, 0, stream>>>(src, dst, dinv, bufA, bufB, E, 6);

    // ---- scatter-mean readout over graphs
    k_readout_zero   <<<dim3((GCN_G * GCN_OUT + TB - 1) / TB), blk, 0, stream>>>(gsum, cnt);
    k_readout_scatter<<<dim3((N * OUTF + TB - 1) / TB), blk, 0, stream>>>(bufB, batch, gsum, cnt, N);
    k_readout_final  <<<dim3((GCN_G * GCN_OUT + TB - 1) / TB), blk, 0, stream>>>(gsum, cnt, (float*)d_out);
}